// StructuralGCN_21586505630435
// MI455X (gfx1250) — compile-verified
//
#include <hip/hip_runtime.h>
#include <hip/hip_bf16.h>
#include <math.h>

typedef float v2f __attribute__((ext_vector_type(2)));
typedef float v8f __attribute__((ext_vector_type(8)));

#define N_NODES 100000
#define N_EDGES 1600000
#define IN_DIM  128
#define HID     64
#define N_CLOSE 5
#define N_GRAPH 512

// ---------------------------------------------------------------- utilities

__global__ void fill_kernel(float* __restrict__ p, float v, int n) {
    int i = blockIdx.x * blockDim.x + threadIdx.x;
    if (i < n) p[i] = v;
}

__device__ __forceinline__ void atomicMaxFloat(float* addr, float val) {
    // sign-split monotone-bits trick: works for any stored value incl. -inf
    if (val >= 0.0f) {
        atomicMax((int*)addr, __float_as_int(val));
    } else {
        atomicMin((unsigned int*)addr, __float_as_uint(val));
    }
}

// ---------------------------------------------------------------- degree

__global__ void deg_kernel(const int* __restrict__ dst, float* __restrict__ deg, int E) {
    int e = blockIdx.x * blockDim.x + threadIdx.x;
    if (e < E) atomicAdd(&deg[dst[e]], 1.0f);
}

__global__ void rsqrt_kernel(float* __restrict__ d, int n) {
    int i = blockIdx.x * blockDim.x + threadIdx.x;
    if (i < n) d[i] = rsqrtf(d[i] + 1.0f);   // implicit self-loop
}

// ---------------------------------------------------------------- WMMA GEMM
// T[M x 64] = A[M x K] @ W[K x 64], fp32 via V_WMMA_F32_16X16X4_F32.
// Block = 128 threads = 4 waves; block tile = 16 rows x 64 cols;
// wave w owns columns [16w, 16w+16). A tile + full W staged in LDS.

template<int K>
__global__ __launch_bounds__(128)
void gemm_wmma(const float* __restrict__ A, const float* __restrict__ W,
               float* __restrict__ T, int M) {
    constexpr int AP = K + 4;   // padded A row stride (bank-conflict free frags)
    constexpr int WP = 80;      // padded W row stride (half-waves hit disjoint banks)
    __shared__ float As[16 * AP];
    __shared__ float Ws[K * WP];

    const int tid     = threadIdx.x;
    const int rowBase = blockIdx.x * 16;

    // cooperative, coalesced global -> LDS staging
    for (int i = tid; i < 16 * K; i += 128) {
        int r = i / K, c = i % K;
        int gr = rowBase + r;
        As[r * AP + c] = (gr < M) ? A[(long long)gr * K + c] : 0.0f;
    }
    for (int i = tid; i < K * 64; i += 128) {
        int r = i >> 6, c = i & 63;
        Ws[r * WP + c] = W[i];
    }
    __syncthreads();

    const int lane    = tid & 31;
    const int wave    = tid >> 5;
    const int m       = lane & 15;       // A: M index / B: N index
    const int khalf   = (lane >> 4) * 2; // lanes 16-31 hold K+2, K+3
    const int colBase = wave * 16;

    v8f acc = {};
    #pragma unroll
    for (int k = 0; k < K; k += 4) {
        const int kk = k + khalf;
        v2f a, b;
        a.x = As[m * AP + kk];
        a.y = As[m * AP + kk + 1];
        b.x = Ws[kk * WP + colBase + m];
        b.y = Ws[(kk + 1) * WP + colBase + m];
        acc = __builtin_amdgcn_wmma_f32_16x16x4_f32(
            /*neg_a=*/false, a, /*neg_b=*/false, b,
            /*c_mod=*/(short)0, acc, /*reuse_a=*/false, /*reuse_b=*/false);
    }

    const int col   = colBase + (lane & 15);
    const int rHalf = (lane >> 4) * 8;
    #pragma unroll
    for (int i = 0; i < 8; ++i) {
        int gr = rowBase + rHalf + i;
        if (gr < M) T[(long long)gr * 64 + col] = acc[i];
    }
}

// ---------------------------------------------------------------- edge scatter
// 64 consecutive threads per edge -> coalesced row read of t[src], atomic adds
// into agg[dst] stay resident in the 192MB L2.

__global__ void scatter_kernel(const int* __restrict__ src, const int* __restrict__ dst,
                               const float* __restrict__ dinv,
                               const float* __restrict__ T, float* __restrict__ agg,
                               int total) {
    int idx = blockIdx.x * blockDim.x + threadIdx.x;
    if (idx >= total) return;
    int e = idx >> 6;
    int f = idx & 63;
    int s = src[e], d = dst[e];
    float nrm = dinv[s] * dinv[d];
    atomicAdd(&agg[(long long)d * 64 + f], T[(long long)s * 64 + f] * nrm);
}

// h = relu(agg + t * dinv^2 + b)
__global__ void combine_kernel(const float* __restrict__ agg, const float* __restrict__ T,
                               const float* __restrict__ dinv, const float* __restrict__ b,
                               float* __restrict__ H, int total) {
    int idx = blockIdx.x * blockDim.x + threadIdx.x;
    if (idx >= total) return;
    int i = idx >> 6;
    int f = idx & 63;
    float sn = dinv[i]; sn *= sn;
    float v = agg[idx] + T[idx] * sn + b[f];
    H[idx] = fmaxf(v, 0.0f);
}

// ---------------------------------------------------------------- graph softmax

__global__ void score_kernel(const float* __restrict__ close, const float* __restrict__ Wc,
                             const float* __restrict__ bc, const int* __restrict__ batch,
                             float* __restrict__ sarr, float* __restrict__ smax, int n) {
    int i = blockIdx.x * blockDim.x + threadIdx.x;
    if (i >= n) return;
    float s = bc[0];
    #pragma unroll
    for (int j = 0; j < N_CLOSE; ++j) s += close[i * N_CLOSE + j] * Wc[j];
    sarr[i] = s;
    atomicMaxFloat(&smax[batch[i]], s);
}

__global__ void expsum_kernel(float* __restrict__ sarr, const float* __restrict__ smax,
                              const int* __restrict__ batch,
                              float* __restrict__ ssum, float* __restrict__ counts, int n) {
    int i = blockIdx.x * blockDim.x + threadIdx.x;
    if (i >= n) return;
    int b = batch[i];
    float e = expf(sarr[i] - smax[b]);
    sarr[i] = e;                      // reuse buffer for e
    atomicAdd(&ssum[b], e);
    atomicAdd(&counts[b], 1.0f);
}

// pooled[g][f] = max_i ( e_i / ssum_g * count_g * h[i][f] )
__global__ void pool_kernel(const float* __restrict__ sarr, const float* __restrict__ ssum,
                            const float* __restrict__ counts, const int* __restrict__ batch,
                            const float* __restrict__ H, float* __restrict__ pooled, int total) {
    int idx = blockIdx.x * blockDim.x + threadIdx.x;
    if (idx >= total) return;
    int i = idx >> 6;
    int b = batch[i];
    float w = sarr[i] / ssum[b] * counts[b];
    atomicMaxFloat(&pooled[(long long)b * 64 + (idx & 63)], w * H[idx]);
}

// ---------------------------------------------------------------- final MLP (tiny)

__global__ void mlp_kernel(const float* __restrict__ pooled,
                           const float* __restrict__ Wa1, const float* __restrict__ ba1,
                           const float* __restrict__ Wa2, const float* __restrict__ ba2,
                           float* __restrict__ out, int G) {
    int g = blockIdx.x * blockDim.x + threadIdx.x;
    if (g >= G) return;
    float acc = ba2[0];
    #pragma unroll
    for (int j = 0; j < 16; ++j) {
        float a = ba1[j];
        for (int k = 0; k < 64; ++k) a += pooled[g * 64 + k] * Wa1[k * 16 + j];
        acc += fmaxf(a, 0.0f) * Wa2[j];
    }
    out[g] = acc;
}

// ---------------------------------------------------------------- launcher

extern "C" void kernel_launch(void* const* d_in, const int* in_sizes, int n_in,
                              void* d_out, int out_size, void* d_ws, size_t ws_size,
                              hipStream_t stream) {
    const float* x     = (const float*)d_in[0];
    const int*   ei    = (const int*)  d_in[1];
    const float* close = (const float*)d_in[2];
    const int*   batch = (const int*)  d_in[3];
    const float* W1    = (const float*)d_in[5];
    const float* b1    = (const float*)d_in[6];
    const float* W2    = (const float*)d_in[7];
    const float* b2    = (const float*)d_in[8];
    const float* W3    = (const float*)d_in[9];
    const float* b3    = (const float*)d_in[10];
    const float* Wc    = (const float*)d_in[11];
    const float* bc    = (const float*)d_in[12];
    const float* Wa1   = (const float*)d_in[13];
    const float* ba1   = (const float*)d_in[14];
    const float* Wa2   = (const float*)d_in[15];
    const float* ba2   = (const float*)d_in[16];
    float* out = (float*)d_out;

    const int* src = ei;
    const int* dst = ei + N_EDGES;

    // workspace carve-up (floats)
    float* ws     = (float*)d_ws;
    float* t      = ws;                         // N*64
    float* agg    = t      + (size_t)N_NODES * 64;
    float* h      = agg    + (size_t)N_NODES * 64;
    float* dinv   = h      + (size_t)N_NODES * 64;   // N  (deg -> dinv in place)
    float* sarr   = dinv   + N_NODES;                // N  (score -> exp in place)
    float* smax   = sarr   + N_NODES;                // G
    float* ssum   = smax   + N_GRAPH;                // G
    float* counts = ssum   + N_GRAPH;                // G
    float* pooled = counts + N_GRAPH;                // G*64

    const int NH   = N_NODES * 64;
    const int ET   = N_EDGES * 64;
    const int blkN   = (N_NODES + 255) / 256;
    const int blkNH  = (NH + 255) / 256;
    const int blkET  = (ET + 255) / 256;
    const int blkE   = (N_EDGES + 255) / 256;
    const int gemmBlocks = (N_NODES + 15) / 16;

    // degree -> dinv
    fill_kernel<<<blkN, 256, 0, stream>>>(dinv, 0.0f, N_NODES);
    deg_kernel<<<blkE, 256, 0, stream>>>(dst, dinv, N_EDGES);
    rsqrt_kernel<<<blkN, 256, 0, stream>>>(dinv, N_NODES);

    // ---- layer 1: t = x @ W1 (K=128)
    gemm_wmma<IN_DIM><<<gemmBlocks, 128, 0, stream>>>(x, W1, t, N_NODES);
    fill_kernel<<<blkNH, 256, 0, stream>>>(agg, 0.0f, NH);
    scatter_kernel<<<blkET, 256, 0, stream>>>(src, dst, dinv, t, agg, ET);
    combine_kernel<<<blkNH, 256, 0, stream>>>(agg, t, dinv, b1, h, NH);

    // ---- layer 2: t = h @ W2 (K=64)
    gemm_wmma<HID><<<gemmBlocks, 128, 0, stream>>>(h, W2, t, N_NODES);
    fill_kernel<<<blkNH, 256, 0, stream>>>(agg, 0.0f, NH);
    scatter_kernel<<<blkET, 256, 0, stream>>>(src, dst, dinv, t, agg, ET);
    combine_kernel<<<blkNH, 256, 0, stream>>>(agg, t, dinv, b2, h, NH);

    // ---- layer 3
    gemm_wmma<HID><<<gemmBlocks, 128, 0, stream>>>(h, W3, t, N_NODES);
    fill_kernel<<<blkNH, 256, 0, stream>>>(agg, 0.0f, NH);
    scatter_kernel<<<blkET, 256, 0, stream>>>(src, dst, dinv, t, agg, ET);
    combine_kernel<<<blkNH, 256, 0, stream>>>(agg, t, dinv, b3, h, NH);

    // ---- per-graph softmax weights + weighted max pooling
    fill_kernel<<<(N_GRAPH + 255) / 256, 256, 0, stream>>>(smax, -INFINITY, N_GRAPH);
    fill_kernel<<<(N_GRAPH + 255) / 256, 256, 0, stream>>>(ssum, 0.0f, N_GRAPH);
    fill_kernel<<<(N_GRAPH + 255) / 256, 256, 0, stream>>>(counts, 0.0f, N_GRAPH);
    fill_kernel<<<(N_GRAPH * 64 + 255) / 256, 256, 0, stream>>>(pooled, -INFINITY, N_GRAPH * 64);

    score_kernel<<<blkN, 256, 0, stream>>>(close, Wc, bc, batch, sarr, smax, N_NODES);
    expsum_kernel<<<blkN, 256, 0, stream>>>(sarr, smax, batch, ssum, counts, N_NODES);
    pool_kernel<<<blkNH, 256, 0, stream>>>(sarr, ssum, counts, batch, h, pooled, NH);

    // ---- tiny per-graph MLP -> d_out [G,1]
    mlp_kernel<<<(N_GRAPH + 255) / 256, 256, 0, stream>>>(pooled, Wa1, ba1, Wa2, ba2, out, N_GRAPH);
}